// LinearAttention_49804440764634
// MI455X (gfx1250) — compile-verified
//
#include <hip/hip_runtime.h>

// ---------------------------------------------------------------------------
// Causal linear attention (chunkwise) for MI455X / gfx1250, fp32 WMMA path.
// Shapes: N=4, L=4096, H=16, D=M=64, CHUNK=128, G=32 chunks.
// 3 passes: per-chunk KV state -> exclusive chunk prefix -> per-chunk output.
// Matmuls on V_WMMA_F32_16X16X4_F32; untransformed operands staged with
// GLOBAL_LOAD_ASYNC_TO_LDS_B128 (ASYNCcnt), phi(q)/phi(k) via VGPR path.
// ---------------------------------------------------------------------------

typedef float v2f __attribute__((ext_vector_type(2)));
typedef float v8f __attribute__((ext_vector_type(8)));

#define NB   4
#define LL   4096
#define HH   16
#define DD   64
#define CC   128
#define GG   32
#define ROWSTRIDE 1024          // H*D elements between consecutive l
#define SQK 68                  // LDS stride (floats), 68%64==4 -> conflict-free
#define SSS 132                 // LDS stride for 128x128 score tile
#define SKVS 68                 // LDS stride for 64x64 KV state

__device__ __forceinline__ float phi(float x) {            // elu(x)+1
    return x > 0.0f ? x + 1.0f : __expf(x);
}
__device__ __forceinline__ float4 phi4(float4 v) {
    v.x = phi(v.x); v.y = phi(v.y); v.z = phi(v.z); v.w = phi(v.w);
    return v;
}

// Async memory -> LDS copy of 16 bytes per lane (ASYNCcnt-tracked).
// ldsOff: byte offset within this workgroup's LDS allocation (16B aligned).
__device__ __forceinline__ void async_g2lds_b128(unsigned ldsOff, const void* g) {
    asm volatile("global_load_async_to_lds_b128 %0, %1, off"
                 :: "v"(ldsOff), "v"(g) : "memory");
}
__device__ __forceinline__ void wait_async0() {
    asm volatile("s_wait_asynccnt 0x0" ::: "memory");
}

__device__ __forceinline__ v8f wmma4(v2f a, v2f b, v8f c) {
    // D(16x16,f32) = A(16x4,f32) x B(4x16,f32) + C
    return __builtin_amdgcn_wmma_f32_16x16x4_f32(false, a, false, b,
                                                 (short)0, c, false, false);
}

// Fragment loaders for the 16x16x4 f32 operand layouts (wave32):
//   A: lane ln holds row m=ln; VGPR j holds K = k0 + 2*hl + j   (hl = lane>>4)
//   B: lane ln holds col n=ln; VGPR j holds K = k0 + 2*hl + j
// L1: matrix stored row-major as S[row][col] with (row=m|n, col=k)
__device__ __forceinline__ v2f ldsL1(const float* s, int stride, int r0, int k0,
                                     int ln, int hl) {
    v2f f;
    const float* p = s + (r0 + ln) * stride + k0 + 2 * hl;
    f.x = p[0]; f.y = p[1];
    return f;
}
// L2: matrix stored row-major as S[k][col] (k varies with VGPR/half-wave)
__device__ __forceinline__ v2f ldsL2(const float* s, int stride, int k0, int c0,
                                     int ln, int hl) {
    v2f f;
    f.x = s[(k0 + 2 * hl + 0) * stride + c0 + ln];
    f.y = s[(k0 + 2 * hl + 1) * stride + c0 + ln];
    return f;
}

// ---------------------------------------------------------------------------
// Pass 1: per-chunk state.  block = (n,g,h).  kv = phi(k)^T x v  (64x64),
// ksum = column sums of phi(k).  8 waves: each owns 2 of 16 kv tiles.
// ---------------------------------------------------------------------------
__global__ void __launch_bounds__(256) k_chunk_state(
        const float* __restrict__ K, const float* __restrict__ V,
        float* __restrict__ wkv, float* __restrict__ wks) {
    extern __shared__ float smem[];
    float* sK = smem;                 // 128 x SQK
    float* sV = sK + CC * SQK;        // 128 x SQK
    const unsigned sVoff = (unsigned)(CC * SQK) * 4u;   // LDS byte offset of sV

    const int tid = threadIdx.x, wave = tid >> 5, lane = tid & 31;
    const int ln = lane & 15, hl = lane >> 4;
    const int b = blockIdx.x;                     // ((n*32+g)*16+h)
    const int h = b & 15, ng = b >> 4, g = ng & 31, n = ng >> 5;
    const long base = (((long)n * LL + (long)g * CC) * HH + h) * DD;

    // V: async memory->LDS (no transform needed); K: VGPR path with phi
    #pragma unroll
    for (int i = 0; i < 8; ++i) {
        int idx = tid + i * 256;
        int c = idx >> 4, c4 = (idx & 15) * 4;
        async_g2lds_b128(sVoff + (unsigned)(c * SQK + c4) * 4u,
                         V + base + (long)c * ROWSTRIDE + c4);
        float4 vk = *(const float4*)(K + base + (long)c * ROWSTRIDE + c4);
        *(float4*)&sK[c * SQK + c4] = phi4(vk);
    }
    wait_async0();
    __syncthreads();

    v8f acc[2];
    #pragma unroll
    for (int t = 0; t < 2; ++t)
        #pragma unroll
        for (int r = 0; r < 8; ++r) acc[t][r] = 0.0f;

    #pragma unroll
    for (int t = 0; t < 2; ++t) {
        int tile = 2 * wave + t, dt = tile >> 2, mt = tile & 3;
        #pragma unroll
        for (int k0 = 0; k0 < CC; k0 += 4) {
            v2f a  = ldsL2(sK, SQK, k0, 16 * dt, ln, hl);  // A[d][c] = kc[c][d]
            v2f bb = ldsL2(sV, SQK, k0, 16 * mt, ln, hl);  // B[c][m] = vc[c][m]
            acc[t] = wmma4(a, bb, acc[t]);
        }
    }

    if (tid < DD) {                      // ksum over the chunk
        float s = 0.0f;
        for (int c = 0; c < CC; ++c) s += sK[c * SQK + tid];
        wks[(long)b * DD + tid] = s;
    }

    float* okv = wkv + (long)b * (DD * DD);
    #pragma unroll
    for (int t = 0; t < 2; ++t) {
        int tile = 2 * wave + t, dt = tile >> 2, mt = tile & 3;
        #pragma unroll
        for (int r = 0; r < 8; ++r)
            okv[(16 * dt + r + 8 * hl) * DD + 16 * mt + ln] = acc[t][r];
    }
}

// ---------------------------------------------------------------------------
// Pass 2: in-place exclusive prefix over chunks.  block = (n,h), 64 blocks.
// ---------------------------------------------------------------------------
__global__ void __launch_bounds__(256) k_prefix(
        float* __restrict__ wkv, float* __restrict__ wks) {
    const int tid = threadIdx.x, b = blockIdx.x;
    const int n = b >> 4, h = b & 15;
    float acc[16];
    #pragma unroll
    for (int j = 0; j < 16; ++j) acc[j] = 0.0f;
    float accs = 0.0f;
    for (int g = 0; g < GG; ++g) {
        long idx = (long)(n * GG + g) * HH + h;
        float* p = wkv + idx * (DD * DD);
        #pragma unroll
        for (int j = 0; j < 16; ++j) {
            int e = tid + j * 256;
            float v = p[e];
            p[e] = acc[j];
            acc[j] += v;
        }
        if (tid < DD) {
            float* ps = wks + idx * DD;
            float v = ps[tid];
            ps[tid] = accs;
            accs += v;
        }
    }
}

// ---------------------------------------------------------------------------
// Pass 3: outputs.  block = (n,g,h).
//   out = phi(q) x kv_prev  +  tril(phi(q) phi(k)^T) x v
//   z   = phi(q).ksum_prev  +  rowsum(masked scores)  + eps
// Wave w owns score/output row-tile w (rows 16w..16w+15); causal tiles only.
// ---------------------------------------------------------------------------
__global__ void __launch_bounds__(256) k_output(
        const float* __restrict__ Q, const float* __restrict__ K,
        const float* __restrict__ V, const float* __restrict__ wkv,
        const float* __restrict__ wks, float* __restrict__ O) {
    extern __shared__ float smem[];
    float* sQ  = smem;                  // 128 x SQK
    float* sK  = sQ + CC * SQK;         // 128 x SQK
    float* sV  = sK + CC * SQK;         // 128 x SQK
    float* sS  = sV + CC * SQK;         // 128 x SSS  (masked scores)
    float* sKV = sS + CC * SSS;         // 64 x SKVS  (exclusive-prefix kv)
    float* sZ  = sKV + DD * SKVS;       // 128
    float* sKs = sZ + CC;               // 64
    const unsigned sVoff  = (unsigned)(2 * CC * SQK) * 4u;
    const unsigned sKVoff = (unsigned)(3 * CC * SQK + CC * SSS) * 4u;

    const int tid = threadIdx.x, wave = tid >> 5, lane = tid & 31;
    const int ln = lane & 15, hl = lane >> 4;
    const int b = blockIdx.x;
    const int h = b & 15, ng = b >> 4, g = ng & 31, n = ng >> 5;
    const long base = (((long)n * LL + (long)g * CC) * HH + h) * DD;

    // V + prefix-KV: async memory->LDS; Q/K: VGPR path with phi
    const float* ikv = wkv + (long)b * (DD * DD);
    #pragma unroll
    for (int i = 0; i < 4; ++i) {
        int idx = tid + i * 256;                   // 1024 float4 total
        int r = idx >> 4, c4 = (idx & 15) * 4;
        async_g2lds_b128(sKVoff + (unsigned)(r * SKVS + c4) * 4u,
                         ikv + r * DD + c4);
    }
    #pragma unroll
    for (int i = 0; i < 8; ++i) {
        int idx = tid + i * 256;
        int c = idx >> 4, c4 = (idx & 15) * 4;
        async_g2lds_b128(sVoff + (unsigned)(c * SQK + c4) * 4u,
                         V + base + (long)c * ROWSTRIDE + c4);
        float4 vq = *(const float4*)(Q + base + (long)c * ROWSTRIDE + c4);
        float4 vk = *(const float4*)(K + base + (long)c * ROWSTRIDE + c4);
        *(float4*)&sQ[c * SQK + c4] = phi4(vq);
        *(float4*)&sK[c * SQK + c4] = phi4(vk);
    }
    if (tid < DD) sKs[tid] = wks[(long)b * DD + tid];
    wait_async0();
    __syncthreads();

    // normalizer, inter-chunk part: z = phi(q_l) . ksum_prev
    if (tid < CC) {
        float acc = 0.0f;
        for (int d = 0; d < DD; ++d) acc += sQ[tid * SQK + d] * sKs[d];
        sZ[tid] = acc + 1e-6f;
    }

    // inter-chunk output: out = phi(q) x kv_prev   (wave w -> row-tile w)
    v8f out_acc[4];
    #pragma unroll
    for (int t = 0; t < 4; ++t)
        #pragma unroll
        for (int r = 0; r < 8; ++r) out_acc[t][r] = 0.0f;
    #pragma unroll
    for (int nt = 0; nt < 4; ++nt) {
        #pragma unroll
        for (int k0 = 0; k0 < DD; k0 += 4) {
            v2f a  = ldsL1(sQ, SQK, 16 * wave, k0, ln, hl);
            v2f bb = ldsL2(sKV, SKVS, k0, 16 * nt, ln, hl);
            out_acc[nt] = wmma4(a, bb, out_acc[nt]);
        }
    }

    // scores: S = phi(q) phi(k)^T, causal tiles only; diagonal tile masked
    for (int nt = 0; nt <= wave; ++nt) {
        v8f sacc;
        #pragma unroll
        for (int r = 0; r < 8; ++r) sacc[r] = 0.0f;
        #pragma unroll
        for (int k0 = 0; k0 < DD; k0 += 4) {
            v2f a  = ldsL1(sQ, SQK, 16 * wave, k0, ln, hl);
            v2f bb = ldsL1(sK, SQK, 16 * nt, k0, ln, hl); // B[k][n] = kc[n][k]
            sacc = wmma4(a, bb, sacc);
        }
        int nn = 16 * nt + ln;
        #pragma unroll
        for (int r = 0; r < 8; ++r) {
            int m = 16 * wave + r + 8 * hl;
            float val = sacc[r];
            if (nt == wave && nn > m) val = 0.0f;   // causal mask on diagonal
            sS[m * SSS + nn] = val;
        }
    }
    __syncthreads();

    // normalizer, intra part: rowsum of masked scores (cols 0..t only)
    if (tid < CC) {
        float acc = 0.0f;
        for (int j = 0; j <= tid; ++j) acc += sS[tid * SSS + j];
        sZ[tid] += acc;
    }

    // intra-chunk output: out += S x v   (contract only causal k-tiles)
    #pragma unroll
    for (int nt = 0; nt < 4; ++nt) {
        for (int k0 = 0; k0 < 16 * (wave + 1); k0 += 4) {
            v2f a  = ldsL1(sS, SSS, 16 * wave, k0, ln, hl);
            v2f bb = ldsL2(sV, SQK, k0, 16 * nt, ln, hl);
            out_acc[nt] = wmma4(a, bb, out_acc[nt]);
        }
    }
    __syncthreads();   // sZ final before use

    float* go = O + base;
    #pragma unroll
    for (int nt = 0; nt < 4; ++nt) {
        int mv = 16 * nt + ln;
        #pragma unroll
        for (int r = 0; r < 8; ++r) {
            int m = 16 * wave + r + 8 * hl;
            go[(long)m * ROWSTRIDE + mv] = out_acc[nt][r] / sZ[m];
        }
    }
}

// ---------------------------------------------------------------------------
extern "C" void kernel_launch(void* const* d_in, const int* in_sizes, int n_in,
                              void* d_out, int out_size, void* d_ws, size_t ws_size,
                              hipStream_t stream) {
    const float* Q = (const float*)d_in[0];
    const float* K = (const float*)d_in[1];
    const float* V = (const float*)d_in[2];
    float* O = (float*)d_out;

    const int nChunks = NB * GG * HH;                 // 2048
    float* wkv = (float*)d_ws;                        // 2048 * 4096 f32
    float* wks = wkv + (size_t)nChunks * DD * DD;     // 2048 * 64 f32

    const size_t lds1 = (size_t)(2 * CC * SQK) * sizeof(float);
    const size_t lds3 = (size_t)(3 * CC * SQK + CC * SSS + DD * SKVS + CC + DD)
                        * sizeof(float);

    hipFuncSetAttribute((const void*)k_chunk_state,
                        hipFuncAttributeMaxDynamicSharedMemorySize, (int)lds1);
    hipFuncSetAttribute((const void*)k_output,
                        hipFuncAttributeMaxDynamicSharedMemorySize, (int)lds3);

    k_chunk_state<<<nChunks, 256, lds1, stream>>>(K, V, wkv, wks);
    k_prefix<<<NB * HH, 256, 0, stream>>>(wkv, wks);
    k_output<<<nChunks, 256, lds3, stream>>>(Q, K, V, wkv, wks, O);
}